// local_NCC_2d_1760936591436
// MI455X (gfx1250) — compile-verified
//
#include <hip/hip_runtime.h>

// ---------------------------------------------------------------------------
// LNCC registration loss for MI455X (gfx1250), fp32, bandwidth-shaped.
//   128 images of 384x384 bilinearly warped, mean template, 3x3-box LNCC,
//   + l2 grad penalty on the flows. Output: 1 float scalar.
// ~600 MB HBM -> ~26 us at 23.3 TB/s; compute is negligible. This revision:
//   * B128 (float4) traffic in the warp/template/grad streaming passes
//   * GLOBAL_LOAD_ASYNC_TO_LDS_B32 (ASYNCcnt) for the LNCC halo tiles
//   * V_WMMA_F32_16X16X4_F32 ones-matrix wave reductions (deterministic,
//     no atomics anywhere)
// Workspace: (128*147456 + 147456 + 18432 + 4096) * 4 B ~= 76.2 MB.
// ---------------------------------------------------------------------------

namespace {
constexpr int NIMG = 128;            // B*G
constexpr int H    = 384;
constexpr int W    = 384;
constexpr int HWc  = H * W;          // 147456
constexpr int TILE = 32;
constexpr int GX   = W / TILE;       // 12
constexpr int GY   = H / TILE;       // 12
constexpr int NCC_BLOCKS  = GX * GY * NIMG;     // 18432
constexpr int GRAD_BLOCKS = 4096;
constexpr int WARP_TOTAL  = NIMG * 2 * HWc;     // 37,748,736 flow elements
constexpr float SMOOTH_DR = 1e-5f;
}

typedef float v2f __attribute__((ext_vector_type(2)));
typedef float v4f __attribute__((ext_vector_type(4)));
typedef float v8f __attribute__((ext_vector_type(8)));

// ---------------------------------------------------------------------------
// Async global->LDS copy, one f32 per active lane (GV mode, per-lane 64-bit
// global address, per-lane LDS byte offset). Tracked by ASYNCcnt.
// ---------------------------------------------------------------------------
__device__ __forceinline__ void async_load_f32_to_lds(unsigned lds_off,
                                                      const float* gptr) {
  asm volatile("global_load_async_to_lds_b32 %0, %1, off"
               :: "v"(lds_off), "v"((unsigned long long)(uintptr_t)gptr)
               : "memory");
}
__device__ __forceinline__ void wait_asynccnt0() {
  asm volatile("s_wait_asynccnt 0x0" ::: "memory");
}

// ---------------------------------------------------------------------------
// Sum of a float across all 32 lanes via V_WMMA_F32_16X16X4_F32.
// A is 16x4 fp32 = 64 cells = 32 lanes x 2 VGPRs (a bijection whatever the
// striping), B = ones(4x16)  =>  D[m][n] = rowsum_m for every n. Per the
// documented C/D layout, lanes 0-15 hold rows 0-7 in c[0..7] and lanes 16-31
// hold rows 8-15, so sum(c[0..7]) + shfl_xor(.,16) = full wave sum.
// Requires EXEC all ones: call only from fully-converged code.
// ---------------------------------------------------------------------------
__device__ __forceinline__ float wave_sum_wmma(float p) {
  v2f a; a[0] = p;    a[1] = 0.0f;
  v2f b; b[0] = 1.0f; b[1] = 1.0f;
  v8f c = {};
  c = __builtin_amdgcn_wmma_f32_16x16x4_f32(
      /*neg_a=*/false, a, /*neg_b=*/false, b,
      /*c_mod=*/(short)0, c, /*reuse_a=*/false, /*reuse_b=*/false);
  float t = c[0] + c[1] + c[2] + c[3] + c[4] + c[5] + c[6] + c[7];
  t += __shfl_xor(t, 16, 32);
  return t;  // full 32-lane sum, present in every lane
}

// Deterministic 256-thread block sum (8 waves). Valid in thread 0.
__device__ __forceinline__ float block_sum256(float p, float* lds8) {
  float wsum = wave_sum_wmma(p);
  const int tid = threadIdx.x;
  if ((tid & 31) == 0) lds8[tid >> 5] = wsum;
  __syncthreads();
  float r = 0.0f;
  if (tid == 0) {
#pragma unroll
    for (int i = 0; i < 8; ++i) r += lds8[i];
  }
  return r;
}

// ---------------------------------------------------------------------------
// Pass 1: bilinear warp, border clamp (== map_coordinates order=1 "nearest":
// weights from the unclamped coord, indices clipped). 4 pixels per thread:
// float4 flow loads + float4 store; the 16 image gathers stay scalar.
// ---------------------------------------------------------------------------
__global__ void warp_kernel(const float* __restrict__ images,
                            const float* __restrict__ warps,
                            float* __restrict__ warped) {
  int q = blockIdx.x * blockDim.x + threadIdx.x;   // quad index
  if (q >= NIMG * (HWc / 4)) return;
  int n   = q / (HWc / 4);
  int pq  = q - n * (HWc / 4);
  int pix = pq * 4;                                // x..x+3 in one row
  int y   = pix / W;
  int x0  = pix - y * W;

  const float* img = images + (size_t)n * HWc;
  v4f fx4 = *(const v4f*)(warps + ((size_t)n * 2 + 0) * HWc + pix);
  v4f fy4 = *(const v4f*)(warps + ((size_t)n * 2 + 1) * HWc + pix);

  v4f r;
#pragma unroll
  for (int k = 0; k < 4; ++k) {
    float cx = (float)(x0 + k) + fx4[k];
    float cy = (float)y + fy4[k];
    float flx = floorf(cx), fly = floorf(cy);
    float wx = cx - flx,    wy = cy - fly;
    int xi = (int)flx, yi = (int)fly;
    int x0c = min(max(xi,     0), W - 1);
    int x1c = min(max(xi + 1, 0), W - 1);
    int y0c = min(max(yi,     0), H - 1);
    int y1c = min(max(yi + 1, 0), H - 1);
    float v00 = img[y0c * W + x0c], v01 = img[y0c * W + x1c];
    float v10 = img[y1c * W + x0c], v11 = img[y1c * W + x1c];
    float v0 = v00 + wx * (v01 - v00);
    float v1 = v10 + wx * (v11 - v10);
    r[k] = v0 + wy * (v1 - v0);
  }
  *(v4f*)(warped + (size_t)n * HWc + pix) = r;
}

// ---------------------------------------------------------------------------
// Pass 2: template = mean over the 128 warped images, float4 per thread.
// ---------------------------------------------------------------------------
__global__ void template_kernel(const float* __restrict__ warped,
                                float* __restrict__ tmpl) {
  int q = blockIdx.x * blockDim.x + threadIdx.x;   // quad index over HW/4
  if (q >= HWc / 4) return;
  int pix = q * 4;
  v4f s = {0.0f, 0.0f, 0.0f, 0.0f};
#pragma unroll 4
  for (int n = 0; n < NIMG; ++n)
    s += *(const v4f*)(warped + (size_t)n * HWc + pix);
  *(v4f*)(tmpl + pix) = s * (1.0f / (float)NIMG);
}

// ---------------------------------------------------------------------------
// Pass 3: LNCC. One block = one 32x32 tile of one image. 34x34 halo tiles of
// pred (warped) and target (template) land in LDS via async global->LDS
// loads (EXEC-masked to in-bounds lanes); out-of-image slots are zeroed with
// plain DS stores (== conv SAME zero padding). Each thread does a 2x2 patch.
// ---------------------------------------------------------------------------
__global__ void lncc_kernel(const float* __restrict__ warped,
                            const float* __restrict__ tmpl,
                            float* __restrict__ blk_out) {
  __shared__ float sp[34 * 36];   // pred tile (+1 halo), padded stride 36
  __shared__ float st[34 * 36];   // target tile
  __shared__ float lds8[8];

  const int bx = blockIdx.x, by = blockIdx.y, n = blockIdx.z;
  const int tid = threadIdx.x;
  const int tx0 = bx * TILE - 1;
  const int ty0 = by * TILE - 1;
  const float* wp = warped + (size_t)n * HWc;

  for (int i = tid; i < 34 * 34; i += 256) {
    int ly = i / 34, lx = i - ly * 34;
    int gy = ty0 + ly, gx = tx0 + lx;
    int li = ly * 36 + lx;
    if (gy >= 0 && gy < H && gx >= 0 && gx < W) {
      int gi = gy * W + gx;
      async_load_f32_to_lds((unsigned)(uintptr_t)&sp[li], wp + gi);
      async_load_f32_to_lds((unsigned)(uintptr_t)&st[li], tmpl + gi);
    } else {
      sp[li] = 0.0f;
      st[li] = 0.0f;
    }
  }
  wait_asynccnt0();    // each wave drains its own async copies
  __syncthreads();     // tile visible to all 8 waves

  const int ox = (tid & 15) * 2;   // 2x2 patch per thread
  const int oy = (tid >> 4) * 2;
  float part = 0.0f;
#pragma unroll
  for (int sy = 0; sy < 2; ++sy) {
#pragma unroll
    for (int sx = 0; sx < 2; ++sx) {
      const int cy = oy + sy + 1;
      const int cx = ox + sx + 1;
      float t_s = 0.f, p_s = 0.f, t2 = 0.f, p2 = 0.f, tp = 0.f;
#pragma unroll
      for (int dy = -1; dy <= 1; ++dy) {
#pragma unroll
        for (int dx = -1; dx <= 1; ++dx) {
          float t = st[(cy + dy) * 36 + (cx + dx)];
          float p = sp[(cy + dy) * 36 + (cx + dx)];
          t_s += t; p_s += p;
          t2 += t * t; p2 += p * p; tp += t * p;
        }
      }
      float t_avg = t_s * (1.0f / 9.0f);
      float p_avg = p_s * (1.0f / 9.0f);
      float cross = tp - p_avg * t_s;
      float t_var = fmaxf(t2 - t_avg * t_s, 0.0f);
      float p_var = fmaxf(p2 - p_avg * p_s, 0.0f);
      part += (cross * cross) / (t_var * p_var + SMOOTH_DR);
    }
  }

  float bs = block_sum256(part, lds8);
  if (tid == 0)
    blk_out[((size_t)n * gridDim.y + by) * gridDim.x + bx] = bs;
}

// ---------------------------------------------------------------------------
// Pass 4: flow smoothness partials, float4 per step.
//   dx within a quad is free; the x+3 -> x+4 pair needs one extra scalar;
//   dy is a second float4 load one row down.
// ---------------------------------------------------------------------------
__global__ void grad_kernel(const float* __restrict__ warps,
                            float* __restrict__ blk_out) {
  __shared__ float lds8[8];
  const int tid = threadIdx.x;
  const int nquads = WARP_TOTAL / 4;               // 9,437,184
  const int stride = gridDim.x * blockDim.x;
  float part = 0.0f;
  for (int q = blockIdx.x * blockDim.x + tid; q < nquads; q += stride) {
    int i   = q * 4;                               // flat element index
    int pix = i % HWc;                             // multiple of 4
    int y   = pix / W;
    int x   = pix - y * W;                         // multiple of 4
    v4f v = *(const v4f*)(warps + i);
    float d0 = v[1] - v[0];
    float d1 = v[2] - v[1];
    float d2 = v[3] - v[2];
    part += d0 * d0 + d1 * d1 + d2 * d2;
    if (x < W - 4) {                               // pair (x+3, x+4)
      float rn = warps[i + 4];
      float d3 = rn - v[3];
      part += d3 * d3;
    }
    if (y < H - 1) {
      v4f w = *(const v4f*)(warps + i + W);
#pragma unroll
      for (int k = 0; k < 4; ++k) {
        float d = w[k] - v[k];
        part += d * d;
      }
    }
  }
  float bs = block_sum256(part, lds8);
  if (tid == 0) blk_out[blockIdx.x] = bs;
}

// ---------------------------------------------------------------------------
// Pass 5: deterministic final reduction + scalar combine.
//   loss = -sum(ncc)/(128*HW) + (sum_dx2 + sum_dy2) / (2 * 256*383*384)
// ---------------------------------------------------------------------------
__global__ void finalize_kernel(const float* __restrict__ blk_ncc,
                                const float* __restrict__ blk_grad,
                                float* __restrict__ out) {
  __shared__ float lds8[8];
  const int tid = threadIdx.x;

  float pn = 0.0f;
  for (int i = tid; i < NCC_BLOCKS; i += 256) pn += blk_ncc[i];
  float sn = block_sum256(pn, lds8);
  __syncthreads();   // protect lds8 before reuse

  float pg = 0.0f;
  for (int i = tid; i < GRAD_BLOCKS; i += 256) pg += blk_grad[i];
  float sg = block_sum256(pg, lds8);

  if (tid == 0) {
    const float mean_n = (float)(NIMG * HWc);          // 18,874,368 (exact)
    const float nd2    = 2.0f * 37650432.0f;           // 2 * 256*383*384 (exact)
    out[0] = -(sn / mean_n) + sg / nd2;
  }
}

// ---------------------------------------------------------------------------
extern "C" void kernel_launch(void* const* d_in, const int* in_sizes, int n_in,
                              void* d_out, int out_size, void* d_ws, size_t ws_size,
                              hipStream_t stream) {
  const float* images = (const float*)d_in[0];  // [8,16,1,384,384] f32
  const float* warps  = (const float*)d_in[1];  // [8,16,2,384,384] f32
  float* out = (float*)d_out;                   // scalar

  // Workspace layout (floats). Requires ~76.2 MB.
  float* warped   = (float*)d_ws;                       // NIMG*HW
  float* tmpl     = warped + (size_t)NIMG * HWc;        // HW
  float* blk_ncc  = tmpl + HWc;                         // NCC_BLOCKS
  float* blk_grad = blk_ncc + NCC_BLOCKS;               // GRAD_BLOCKS
  (void)in_sizes; (void)n_in; (void)out_size; (void)ws_size;

  const int warp_quads = NIMG * (HWc / 4);              // 4,718,592
  warp_kernel<<<(warp_quads + 255) / 256, 256, 0, stream>>>(images, warps, warped);

  const int tmpl_quads = HWc / 4;                       // 36,864
  template_kernel<<<(tmpl_quads + 255) / 256, 256, 0, stream>>>(warped, tmpl);

  dim3 g(GX, GY, NIMG);                                 // 12 x 12 x 128
  lncc_kernel<<<g, 256, 0, stream>>>(warped, tmpl, blk_ncc);

  grad_kernel<<<GRAD_BLOCKS, 256, 0, stream>>>(warps, blk_grad);
  finalize_kernel<<<1, 256, 0, stream>>>(blk_ncc, blk_grad, out);
}